// CustomPredictionWrapper_20882130993361
// MI455X (gfx1250) — compile-verified
//
#include <hip/hip_runtime.h>
#include <math.h>

// ---------------------------------------------------------------------------
// Sampling logits modifier for gfx1250 (MI455X).
//   K1: per-row top-k/top-p threshold via histogram select, streamed through a
//       double-buffered async global->LDS pipeline (ASYNCcnt path).
//   K2: streaming elementwise mask + repetition penalty (HBM-bound, dword ops).
// ---------------------------------------------------------------------------

namespace {
constexpr int   B        = 64;
constexpr int   S        = 8;
constexpr int   V        = 50257;
constexpr int   L        = 2048;
constexpr int   ROWS     = B * S;           // 512
constexpr int   TOPK     = 50;
constexpr float TEMP     = 0.8f;
constexpr float TOP_P    = 0.9f;
constexpr float REP_PEN  = 1.2f;
constexpr float NEG_INF  = -1000000000.0f;

// Kernel 1 config
constexpr int T1    = 256;                  // threads / block
constexpr int TILE  = 1024;                 // floats per async tile (4 KB)
constexpr int NTILE = V / TILE;             // 49 full tiles, tail = 81 elems
constexpr int NBIN  = 4096;                 // 12-bit key histogram
constexpr int CAP   = 2048;                 // candidate buffer (power of two)

// Kernel 2 config
constexpr int T2     = 256;
constexpr int CHUNK  = 16384;               // elements per block
constexpr int CHUNKS = (V + CHUNK - 1) / CHUNK;  // 4
constexpr int BMW    = CHUNK / 32;          // presence bitmap words (512)
} // namespace

// Monotonic float -> uint key: descending float order == descending key order.
__device__ __forceinline__ unsigned fkey(float f) {
  unsigned u = __float_as_uint(f);
  return (u & 0x80000000u) ? ~u : (u | 0x80000000u);
}

// ---------------------------------------------------------------------------
// Kernel 1: one block per (b,s) row. Produces cuts[row] = nucleus cutoff in
// scaled (x = logit / TEMP) units. Keep condition downstream: x >= cut.
// ---------------------------------------------------------------------------
__global__ __launch_bounds__(T1)
void topk_cut_kernel(const float* __restrict__ logits, float* __restrict__ cuts) {
  const int row = blockIdx.x;
  const int tid = threadIdx.x;
  const float* __restrict__ src = logits + (size_t)row * V;

  __shared__ unsigned hist[NBIN];     // 16 KB
  __shared__ float    cand[CAP];      //  8 KB
  __shared__ float    tiles[2][TILE]; //  8 KB
  __shared__ unsigned ncand;
  __shared__ int      binStar;

  for (int i = tid; i < NBIN; i += T1) hist[i] = 0u;
  if (tid == 0) { ncand = 0u; binStar = 0; }
  __syncthreads();

  // LDS byte offsets of the two staging buffers (flat-ptr low 32 bits = LDS addr)
  const unsigned ldsA = (unsigned)(size_t)(&tiles[0][0]);
  const unsigned ldsB = (unsigned)(size_t)(&tiles[1][0]);

  // Issue one tile (1024 floats) as 4 async dword loads per lane.
  // GVS addressing: mem = SADDR(row base) + VADDR(byte offset). Dword-aligned
  // always (row bases are only 4B aligned, so b32 is the safe width).
  auto issueTile = [&](int t) {
    const unsigned bufbase  = (t & 1) ? ldsB : ldsA;
    const unsigned goffBase = (unsigned)t * (unsigned)(TILE * 4);
#pragma unroll
    for (int j = 0; j < 4; ++j) {
      const unsigned e   = (unsigned)tid + (unsigned)(j * T1); // element in tile
      const unsigned lds = bufbase + e * 4u;
      const unsigned go  = goffBase + e * 4u;
      asm volatile("global_load_async_to_lds_b32 %0, %1, %2"
                   :: "v"(lds), "v"(go), "s"(src)
                   : "memory");
    }
  };

  // ---- Pass A: 12-bit-key histogram over the row, double-buffered async ----
  issueTile(0);
  for (int t = 0; t < NTILE; ++t) {
    if (t + 1 < NTILE) {
      issueTile(t + 1);
      // 8 per-wave async ops outstanding; wait until only next tile's 4 remain.
      asm volatile("s_wait_asynccnt 0x4" ::: "memory");
    } else {
      asm volatile("s_wait_asynccnt 0x0" ::: "memory");
    }
    __syncthreads();                        // whole tile t visible in LDS
    const float* tp = tiles[t & 1];
#pragma unroll
    for (int j = 0; j < 4; ++j) {
      const float v = tp[tid + j * T1];     // bank-conflict-free dword reads
      atomicAdd(&hist[fkey(v) >> 20], 1u);
    }
    __syncthreads();                        // buf reusable by tile t+2
  }
  // Tail elements [NTILE*TILE, V)
  {
    const int tail0 = NTILE * TILE;
    if (tid < V - tail0) atomicAdd(&hist[fkey(src[tail0 + tid]) >> 20], 1u);
  }
  __syncthreads();

  // ---- Find boundary bin: largest bin b* with count(key-bin >= b*) >= K ----
  if (tid == 0) {
    unsigned acc = 0;
    int bsel = 0;
    for (int bb = NBIN - 1; bb >= 0; --bb) {
      const unsigned c = hist[bb];
      if (acc + c >= (unsigned)TOPK) { bsel = bb; break; }
      acc += c;
    }
    binStar = bsel;
  }
  __syncthreads();

  // ---- Pass B: gather candidates (key-bin >= b*) as SCALED values ----
  {
    const unsigned bs = (unsigned)binStar;
    for (int v = tid; v < V; v += T1) {
      const float l = src[v];
      if ((fkey(l) >> 20) >= bs) {
        const unsigned p = atomicAdd(&ncand, 1u);
        if (p < (unsigned)CAP) cand[p] = l / TEMP;
      }
    }
  }
  __syncthreads();
  const unsigned n = (ncand < (unsigned)CAP) ? ncand : (unsigned)CAP;
  for (int i = tid; i < CAP; i += T1)
    if ((unsigned)i >= n) cand[i] = -3.4e38f;
  __syncthreads();

  // ---- Bitonic sort descending over CAP elements ----
  for (int k = 2; k <= CAP; k <<= 1) {
    for (int j = k >> 1; j > 0; j >>= 1) {
      for (int i = tid; i < CAP; i += T1) {
        const int ixj = i ^ j;
        if (ixj > i) {
          const float a = cand[i], bv = cand[ixj];
          const bool descBlock = ((i & k) == 0);
          if (descBlock ? (a < bv) : (a > bv)) { cand[i] = bv; cand[ixj] = a; }
        }
      }
      __syncthreads();
    }
  }

  // ---- Softmax-cumsum over sorted top-50; nucleus prefix -> cutoff ----
  if (tid == 0) {
    const float m = cand[0];
    float ex[TOPK];
    float sum = 0.f;
#pragma unroll
    for (int i = 0; i < TOPK; ++i) { ex[i] = expf(cand[i] - m); sum += ex[i]; }
    float c = ex[0] / sum;   // cum[0]
    int keep = 1;
    for (int i = 1; i < TOPK; ++i) {
      if (c > TOP_P) break;  // remove[i] = cum[i-1] > P
      keep = i + 1;
      c += ex[i] / sum;
    }
    cuts[row] = cand[keep - 1];
  }
}

// ---------------------------------------------------------------------------
// Kernel 2: streaming mask + repetition penalty. grid = (ROWS, CHUNKS).
// ---------------------------------------------------------------------------
__global__ __launch_bounds__(T2)
void modify_kernel(const float* __restrict__ logits, const int* __restrict__ ids,
                   const float* __restrict__ cuts, float* __restrict__ out) {
  const int row   = blockIdx.x;          // b*S + s
  const int chunk = blockIdx.y;
  const int tid   = threadIdx.x;
  const int b     = row >> 3;            // S == 8
  const int s     = row & 7;
  const int start = chunk * CHUNK;
  const int end   = (start + CHUNK < V) ? (start + CHUNK) : V;
  const bool lastPos = (s == S - 1);

  __shared__ unsigned bitmap[BMW];       // presence bits for this chunk (2 KB)

  if (lastPos) {
    for (int w = tid; w < BMW; w += T2) bitmap[w] = 0u;
    __syncthreads();
    const int* __restrict__ myids = ids + b * L;
    for (int l = tid; l < L; l += T2) {
      const int t = myids[l];
      if (t >= start && t < end) {
        const int o = t - start;
        atomicOr(&bitmap[o >> 5], 1u << (o & 31));
      }
    }
    __syncthreads();
  }

  const float cut = cuts[row];
  const float* __restrict__ sp = logits + (size_t)row * V;
  float* __restrict__ dp = out + (size_t)row * V;

#pragma unroll 4
  for (int v = start + tid; v < end; v += T2) {
    const float x = sp[v] / TEMP;                     // matches jax: logits / 0.8
    float y = (x >= cut) ? x : NEG_INF;
    if (lastPos) {
      const int o = v - start;
      if ((bitmap[o >> 5] >> (o & 31)) & 1u)
        y = (y < 0.f) ? y * REP_PEN : y / REP_PEN;    // penalty AFTER masking
    }
    dp[v] = y;
  }
}

// ---------------------------------------------------------------------------
extern "C" void kernel_launch(void* const* d_in, const int* in_sizes, int n_in,
                              void* d_out, int out_size, void* d_ws, size_t ws_size,
                              hipStream_t stream) {
  (void)in_sizes; (void)n_in; (void)out_size; (void)ws_size;
  const float* logits = (const float*)d_in[0];
  const int*   ids    = (const int*)d_in[1];
  float*       out    = (float*)d_out;
  float*       cuts   = (float*)d_ws;          // ROWS floats (2 KB) of scratch

  topk_cut_kernel<<<dim3(ROWS), dim3(T1), 0, stream>>>(logits, cuts);
  modify_kernel<<<dim3(ROWS, CHUNKS), dim3(T2), 0, stream>>>(logits, ids, cuts, out);
}